// CrfLayer_22428319220205
// MI455X (gfx1250) — compile-verified
//
#include <hip/hip_runtime.h>
#include <hip/hip_bf16.h>

#define B_ 2048
#define T_ 512
#define C_ 32

typedef __attribute__((ext_vector_type(2))) float v2f;
typedef __attribute__((ext_vector_type(8))) float v8f;

#define LOG2E 1.4426950408889634f
#define LN2   0.6931471805599453f

__device__ __forceinline__ float bcastlane(float v, int lane) {
  return __int_as_float(__builtin_amdgcn_readlane(__float_as_int(v), lane));
}

// ---------------------------------------------------------------------------
// Kernel 1: Viterbi forward. One wave per sequence, lane = cur tag.
// max-plus semiring -> pure VALU. All 32 candidates are computed up-front
// (independent v_readlane + v_add), then a 5-level tree argmax with
// first-max tie break shortens the per-step critical path ~6x vs a linear
// scan. Early exit at seq_len (backward kernel starts at len-1).
// ---------------------------------------------------------------------------
__global__ __launch_bounds__(256) void viterbi_fwd(
    const float* __restrict__ logits, const int* __restrict__ seqlen,
    const float* __restrict__ trans, unsigned char* __restrict__ bp,
    int* __restrict__ last_tag) {
  const int lane = threadIdx.x & 31;
  const int wid  = threadIdx.x >> 5;
  const int b    = blockIdx.x * 8 + wid;

  // tc[p] = transitions[p][lane] (column for this lane's cur tag)
  float tc[32];
#pragma unroll
  for (int p = 0; p < 32; ++p) tc[p] = trans[p * C_ + lane];

  const int len = seqlen[b];
  const float* lrow = logits + (size_t)b * T_ * C_;
  float alpha = lrow[lane];

  for (int t = 1; t < len; ++t) {
    float lg = lrow[t * C_ + lane];
    if (t + 1 < len) __builtin_prefetch(lrow + (t + 1) * C_ + lane, 0, 1);

    // all candidates (independent ops)
    float cand[32];
#pragma unroll
    for (int p = 0; p < 32; ++p) cand[p] = bcastlane(alpha, p) + tc[p];

    // 5-level tree argmax; strictly-greater keeps lowest index on ties
    float v16[16]; int i16[16];
#pragma unroll
    for (int j = 0; j < 16; ++j) {
      bool r = cand[2 * j + 1] > cand[2 * j];
      v16[j] = r ? cand[2 * j + 1] : cand[2 * j];
      i16[j] = r ? 2 * j + 1 : 2 * j;
    }
    float v8a[8]; int i8a[8];
#pragma unroll
    for (int j = 0; j < 8; ++j) {
      bool r = v16[2 * j + 1] > v16[2 * j];
      v8a[j] = r ? v16[2 * j + 1] : v16[2 * j];
      i8a[j] = r ? i16[2 * j + 1] : i16[2 * j];
    }
    float v4[4]; int i4[4];
#pragma unroll
    for (int j = 0; j < 4; ++j) {
      bool r = v8a[2 * j + 1] > v8a[2 * j];
      v4[j] = r ? v8a[2 * j + 1] : v8a[2 * j];
      i4[j] = r ? i8a[2 * j + 1] : i8a[2 * j];
    }
    float v2a[2]; int i2a[2];
#pragma unroll
    for (int j = 0; j < 2; ++j) {
      bool r = v4[2 * j + 1] > v4[2 * j];
      v2a[j] = r ? v4[2 * j + 1] : v4[2 * j];
      i2a[j] = r ? i4[2 * j + 1] : i4[2 * j];
    }
    bool r = v2a[1] > v2a[0];
    float best = r ? v2a[1] : v2a[0];
    int  bestp = r ? i2a[1] : i2a[0];

    alpha = lg + best;
    bp[((size_t)b * T_ + t) * C_ + lane] = (unsigned char)bestp;
  }

  // wave argmax of final state, lowest-index tie break
  float bv = alpha; int bi = lane;
#pragma unroll
  for (int m = 16; m >= 1; m >>= 1) {
    float ov = __shfl_xor(bv, m, 32);
    int   oi = __shfl_xor(bi, m, 32);
    if (ov > bv || (ov == bv && oi < bi)) { bv = ov; bi = oi; }
  }
  if (lane == 0) last_tag[b] = bi;
}

// ---------------------------------------------------------------------------
// Kernel 2: forward algorithm (log partition) with V_WMMA_F32_16X16X4_F32.
// One wave handles 16 sequences; alpha lives in WMMA C/D layout:
//   acc{m}[r] @ lane l : cur tag = r + 8*(l>=16) + 16*m, batch = (l & 15).
// Constant A = exp(T)^T preloaded in A-layout (M=cur, K=prev).
// Per step: m = rowmax (tree); e = exp2((alpha-m)*log2e) staged to LDS
// [prev][batch]; reloaded in B-layout; 2 M-tiles x (two 4-deep WMMA chains);
// alpha' = fma(log2(sum), ln2, m) + logit. Raw v_exp/v_log (args never
// denormal: sum >= min(exp(T)) > 0.01) -> no fixup VALU on the chain.
// ---------------------------------------------------------------------------
__global__ __launch_bounds__(64) void crf_forward_wmma(
    const float* __restrict__ logits, const int* __restrict__ seqlen,
    const float* __restrict__ trans, float* __restrict__ logZ) {
  __shared__ float estage[2][C_ * 16];  // per-wave staging: [prev tag][batch-in-wave]
  const int lane = threadIdx.x & 31;
  const int wid  = threadIdx.x >> 5;
  const int bb   = lane & 15;   // batch index within wave (N)
  const int h    = lane >> 4;   // lane half
  const int b    = blockIdx.x * 32 + wid * 16 + bb;
  float* es = estage[wid];

  // A tiles (constant): a{m}[k] component j = exp(T[4k+2h+j][16m+bb])
  v2f a0[8], a1[8];
#pragma unroll
  for (int k = 0; k < 8; ++k) {
    const int p0 = 4 * k + 2 * h;
    a0[k].x = __builtin_amdgcn_exp2f(trans[(p0 + 0) * C_ + bb] * LOG2E);
    a0[k].y = __builtin_amdgcn_exp2f(trans[(p0 + 1) * C_ + bb] * LOG2E);
    a1[k].x = __builtin_amdgcn_exp2f(trans[(p0 + 0) * C_ + 16 + bb] * LOG2E);
    a1[k].y = __builtin_amdgcn_exp2f(trans[(p0 + 1) * C_ + 16 + bb] * LOG2E);
  }

  const int len = seqlen[b];
  int maxlen = len;
#pragma unroll
  for (int m = 16; m >= 1; m >>= 1) {
    int o = __shfl_xor(maxlen, m, 32);
    maxlen = (o > maxlen) ? o : maxlen;
  }

  const float* lptr = logits + (size_t)b * T_ * C_;
  const int c0 = 8 * h;  // column base for this lane half

  // init alpha = logits[:, 0, :] in C/D layout
  float acc0[8], acc1[8];
  {
    float4 x0 = ((const float4*)(lptr + c0))[0];
    float4 x1 = ((const float4*)(lptr + c0))[1];
    float4 y0 = ((const float4*)(lptr + 16 + c0))[0];
    float4 y1 = ((const float4*)(lptr + 16 + c0))[1];
    acc0[0]=x0.x; acc0[1]=x0.y; acc0[2]=x0.z; acc0[3]=x0.w;
    acc0[4]=x1.x; acc0[5]=x1.y; acc0[6]=x1.z; acc0[7]=x1.w;
    acc1[0]=y0.x; acc1[1]=y0.y; acc1[2]=y0.z; acc1[3]=y0.w;
    acc1[4]=y1.x; acc1[5]=y1.y; acc1[6]=y1.z; acc1[7]=y1.w;
  }

  for (int t = 1; t < maxlen; ++t) {
    const float* lt = lptr + t * C_;
    if (t + 1 < maxlen) __builtin_prefetch(lt + C_, 0, 1);
    float4 L00 = ((const float4*)(lt + c0))[0];
    float4 L01 = ((const float4*)(lt + c0))[1];
    float4 L10 = ((const float4*)(lt + 16 + c0))[0];
    float4 L11 = ((const float4*)(lt + 16 + c0))[1];
    float lg0[8] = {L00.x, L00.y, L00.z, L00.w, L01.x, L01.y, L01.z, L01.w};
    float lg1[8] = {L10.x, L10.y, L10.z, L10.w, L11.x, L11.y, L11.z, L11.w};

    // per-batch max over all 32 tags: 4-level tree over regs + cross-half
    float m8[8];
#pragma unroll
    for (int r = 0; r < 8; ++r) m8[r] = fmaxf(acc0[r], acc1[r]);
    float m4[4];
#pragma unroll
    for (int r = 0; r < 4; ++r) m4[r] = fmaxf(m8[2 * r], m8[2 * r + 1]);
    float mx = fmaxf(fmaxf(m4[0], m4[1]), fmaxf(m4[2], m4[3]));
    mx = fmaxf(mx, __shfl_xor(mx, 16, 32));

    // e = exp(alpha - m) staged to LDS at [prev][bb]  (raw v_exp_f32)
#pragma unroll
    for (int r = 0; r < 8; ++r) {
      es[(c0 + r) * 16 + bb]      = __builtin_amdgcn_exp2f((acc0[r] - mx) * LOG2E);
      es[(16 + c0 + r) * 16 + bb] = __builtin_amdgcn_exp2f((acc1[r] - mx) * LOG2E);
    }
    asm volatile("s_wait_dscnt 0" ::: "memory");  // same-wave DS handoff

    // 4 independent 4-deep WMMA chains (2 per M-tile)
    v8f d0a = {0.f,0.f,0.f,0.f,0.f,0.f,0.f,0.f};
    v8f d0b = {0.f,0.f,0.f,0.f,0.f,0.f,0.f,0.f};
    v8f d1a = {0.f,0.f,0.f,0.f,0.f,0.f,0.f,0.f};
    v8f d1b = {0.f,0.f,0.f,0.f,0.f,0.f,0.f,0.f};
#pragma unroll
    for (int k = 0; k < 4; ++k) {
      v2f bmA;  // B-layout: K = 4k + 2h + j, N = bb
      bmA.x = es[(4 * k + 2 * h + 0) * 16 + bb];
      bmA.y = es[(4 * k + 2 * h + 1) * 16 + bb];
      v2f bmB;
      bmB.x = es[(4 * (k + 4) + 2 * h + 0) * 16 + bb];
      bmB.y = es[(4 * (k + 4) + 2 * h + 1) * 16 + bb];
      d0a = __builtin_amdgcn_wmma_f32_16x16x4_f32(false, a0[k], false, bmA,
                                                  (short)0, d0a, false, false);
      d1a = __builtin_amdgcn_wmma_f32_16x16x4_f32(false, a1[k], false, bmA,
                                                  (short)0, d1a, false, false);
      d0b = __builtin_amdgcn_wmma_f32_16x16x4_f32(false, a0[k + 4], false, bmB,
                                                  (short)0, d0b, false, false);
      d1b = __builtin_amdgcn_wmma_f32_16x16x4_f32(false, a1[k + 4], false, bmB,
                                                  (short)0, d1b, false, false);
    }

    const bool active = (t < len);
#pragma unroll
    for (int r = 0; r < 8; ++r) {
      float n0 = fmaf(__builtin_amdgcn_logf(d0a[r] + d0b[r]), LN2, mx) + lg0[r];
      float n1 = fmaf(__builtin_amdgcn_logf(d1a[r] + d1b[r]), LN2, mx) + lg1[r];
      acc0[r] = active ? n0 : acc0[r];
      acc1[r] = active ? n1 : acc1[r];
    }
  }

  // log_norm = logsumexp over tags of final alpha
  float m8[8];
#pragma unroll
  for (int r = 0; r < 8; ++r) m8[r] = fmaxf(acc0[r], acc1[r]);
  float mx = fmaxf(fmaxf(fmaxf(m8[0], m8[1]), fmaxf(m8[2], m8[3])),
                   fmaxf(fmaxf(m8[4], m8[5]), fmaxf(m8[6], m8[7])));
  mx = fmaxf(mx, __shfl_xor(mx, 16, 32));
  float s = 0.f;
#pragma unroll
  for (int r = 0; r < 8; ++r)
    s += __builtin_amdgcn_exp2f((acc0[r] - mx) * LOG2E) +
         __builtin_amdgcn_exp2f((acc1[r] - mx) * LOG2E);
  s += __shfl_xor(s, 16, 32);
  if (lane < 16) logZ[b] = fmaf(__builtin_amdgcn_logf(s), LN2, mx);
}

// ---------------------------------------------------------------------------
// Kernel 3: gold path score (unary + binary). One wave per sequence.
// ---------------------------------------------------------------------------
__global__ __launch_bounds__(256) void gold_score(
    const float* __restrict__ logits, const int* __restrict__ seqlen,
    const int* __restrict__ labels, const float* __restrict__ trans,
    float* __restrict__ gold) {
  const int lane = threadIdx.x & 31;
  const int wid  = threadIdx.x >> 5;
  const int b    = blockIdx.x * 8 + wid;
  const int len  = seqlen[b];
  const float* lrow = logits + (size_t)b * T_ * C_;
  const int* lab = labels + (size_t)b * T_;
  float acc = 0.f;
  for (int t = lane; t < len; t += 32) {
    int l0 = lab[t]; l0 = (l0 > 0) ? l0 : 0;
    acc += lrow[t * C_ + l0];
    if (t < len - 1) {
      int l1 = lab[t + 1]; l1 = (l1 > 0) ? l1 : 0;
      acc += trans[l0 * C_ + l1];
    }
  }
#pragma unroll
  for (int m = 16; m >= 1; m >>= 1) acc += __shfl_xor(acc, m, 32);
  if (lane == 0) gold[b] = acc;
}

// ---------------------------------------------------------------------------
// Kernel 4: Viterbi backward trace. One thread per sequence.
// ---------------------------------------------------------------------------
__global__ __launch_bounds__(256) void viterbi_bwd(
    const int* __restrict__ seqlen, const unsigned char* __restrict__ bp,
    const int* __restrict__ last_tag, float* __restrict__ out_tags) {
  const int b = blockIdx.x * blockDim.x + threadIdx.x;
  if (b >= B_) return;
  const int len = seqlen[b];
  float* orow = out_tags + (size_t)b * T_;
  for (int t = len; t < T_; ++t) orow[t] = 0.0f;  // masked padding
  int tag = last_tag[b];
  for (int t = len - 1; t >= 1; --t) {
    orow[t] = (float)tag;
    tag = (int)bp[((size_t)b * T_ + t) * C_ + tag];
  }
  orow[0] = (float)tag;  // len >= 1 always
}

// ---------------------------------------------------------------------------
// Kernel 5: loss = mean(logZ - gold)
// ---------------------------------------------------------------------------
__global__ __launch_bounds__(256) void loss_reduce(
    const float* __restrict__ logZ, const float* __restrict__ gold,
    float* __restrict__ out_loss) {
  __shared__ float red[256];
  float s = 0.f;
  for (int i = threadIdx.x; i < B_; i += 256) s += logZ[i] - gold[i];
  red[threadIdx.x] = s;
  __syncthreads();
  for (int off = 128; off >= 1; off >>= 1) {
    if ((int)threadIdx.x < off) red[threadIdx.x] += red[threadIdx.x + off];
    __syncthreads();
  }
  if (threadIdx.x == 0) *out_loss = red[0] / (float)B_;
}

extern "C" void kernel_launch(void* const* d_in, const int* in_sizes, int n_in,
                              void* d_out, int out_size, void* d_ws, size_t ws_size,
                              hipStream_t stream) {
  const float* logits = (const float*)d_in[0];
  const int*   seqlen = (const int*)d_in[1];
  const int*   labels = (const int*)d_in[2];
  const float* trans  = (const float*)d_in[3];

  float* out_tags = (float*)d_out;                 // [B, T] decode tags (as float)
  float* out_loss = out_tags + (size_t)B_ * T_;    // scalar loss

  unsigned char* bp  = (unsigned char*)d_ws;       // [B, T, C] backpointers
  size_t bp_bytes    = (size_t)B_ * T_ * C_;
  int*   last_tag    = (int*)(bp + bp_bytes);      // [B]
  float* logZ        = (float*)(last_tag + B_);    // [B]
  float* gold        = (float*)(logZ + B_);        // [B]

  viterbi_fwd     <<<B_ / 8,   256, 0, stream>>>(logits, seqlen, trans, bp, last_tag);
  crf_forward_wmma<<<B_ / 32,  64,  0, stream>>>(logits, seqlen, trans, logZ);
  gold_score      <<<B_ / 8,   256, 0, stream>>>(logits, seqlen, labels, trans, gold);
  viterbi_bwd     <<<B_ / 256, 256, 0, stream>>>(seqlen, bp, last_tag, out_tags);
  loss_reduce     <<<1,        256, 0, stream>>>(logZ, gold, out_loss);
}